// GNN_28003186770423
// MI455X (gfx1250) — compile-verified
//
#include <hip/hip_runtime.h>
#include <hip/hip_bf16.h>

// ---------------------------------------------------------------------------
// GIN (Graph Isomorphism Network) forward, fp32, for gfx1250 (MI455X).
// Matrix work uses V_WMMA_F32_16X16X4_F32 (full fp32 precision; workload is
// bandwidth-bound so there is no payoff for bf16 downcast).
// ---------------------------------------------------------------------------

namespace {
constexpr int   kNodes   = 100000;   // == 3125 * 32, exact multiple of kBR
constexpr int   kEdges   = 600000;
constexpr int   kDim     = 128;
constexpr int   kClasses = 10;
constexpr int   kGraphs  = 512;
constexpr int   kLayers  = 4;
constexpr float kEps     = 1e-5f;

constexpr int kBR  = 32;   // rows per GEMM workgroup
constexpr int kLda = 132;  // padded LDS row stride in floats (528 B, 16B aligned)
}

typedef __attribute__((ext_vector_type(2))) float v2f;
typedef __attribute__((ext_vector_type(8))) float v8f;

// ---------------------------------------------------------------------------
// Utility kernels
// ---------------------------------------------------------------------------
__global__ __launch_bounds__(256)
void gin_fill(float* __restrict__ p, float v, int n) {
    int i = blockIdx.x * 256 + threadIdx.x;
    if (i < n) p[i] = v;
}

__global__ __launch_bounds__(256)
void gin_copy4(float4* __restrict__ dst, const float4* __restrict__ src, int n4) {
    int i = blockIdx.x * 256 + threadIdx.x;
    if (i < n4) dst[i] = src[i];
}

// ---------------------------------------------------------------------------
// Edge scatter-add: z[dst] += h[src], 32 lanes (float4 chunks) per edge
// ---------------------------------------------------------------------------
__global__ __launch_bounds__(256)
void gin_scatter_add(const float* __restrict__ h, const int* __restrict__ src,
                     const int* __restrict__ dst, float* __restrict__ z) {
    long long t = (long long)blockIdx.x * 256 + threadIdx.x;
    if (t >= (long long)kEdges * 32) return;
    int e = (int)(t >> 5);
    int q = ((int)t & 31) << 2;
    int s = src[e];
    int d = dst[e];
    const float4 v = *(const float4*)(h + (size_t)s * kDim + q);
    float* zp = z + (size_t)d * kDim + q;
    atomicAdd(zp + 0, v.x);
    atomicAdd(zp + 1, v.y);
    atomicAdd(zp + 2, v.z);
    atomicAdd(zp + 3, v.w);
}

// ---------------------------------------------------------------------------
// Per-graph sum pooling: pooled[batch[n]] += feat[n]
// ---------------------------------------------------------------------------
__global__ __launch_bounds__(256)
void gin_pool(const float* __restrict__ feat, const int* __restrict__ batch,
              float* __restrict__ pooled) {
    long long t = (long long)blockIdx.x * 256 + threadIdx.x;
    if (t >= (long long)kNodes * 32) return;
    int n = (int)(t >> 5);
    int q = ((int)t & 31) << 2;
    int g = batch[n];
    const float4 v = *(const float4*)(feat + (size_t)n * kDim + q);
    float* pp = pooled + (size_t)g * kDim + q;
    atomicAdd(pp + 0, v.x);
    atomicAdd(pp + 1, v.y);
    atomicAdd(pp + 2, v.z);
    atomicAdd(pp + 3, v.w);
}

// ---------------------------------------------------------------------------
// WMMA GEMM: Out = A @ W + bias, with fused per-column sum / sum-of-squares
// accumulation (for training-mode BatchNorm). In-place capable (Out may
// alias A): each workgroup stages its own 32x128 row block in LDS first.
//
// REQUIRES nrows % 32 == 0 (true here: 100000 = 3125*32), which keeps the
// whole kernel branch-free with EXEC all-1s (WMMA requirement) and lets the
// epilogue use plain strided stores.
//
// Per workgroup: 256 threads = 8 wave32s. Wave w owns columns [16w,16w+16),
// two 16x16 accumulator tiles (rows 0-15 and 16-31 of the block).
// K loop: 32 steps of V_WMMA_F32_16X16X4_F32 per tile.
// ---------------------------------------------------------------------------
__global__ __launch_bounds__(256)
void gin_gemm_stats(const float* A, const float* __restrict__ W,
                    const float* __restrict__ bias, float* Out,
                    float* __restrict__ colsum, float* __restrict__ colsumsq) {
    __shared__ float sA[kBR * kLda];
    const int tid = threadIdx.x;
    const int r0  = blockIdx.x * kBR;

    // Stage A[r0:r0+32, 0:128] into LDS (float4 per thread, 4 iterations).
    #pragma unroll
    for (int i = 0; i < 4; ++i) {
        int idx = tid + 256 * i;      // covers 32 rows x 32 float4-cols
        int row = idx >> 5;
        int c4  = (idx & 31) << 2;
        const float4 v = *(const float4*)(A + (size_t)(r0 + row) * kDim + c4);
        *(float4*)(&sA[row * kLda + c4]) = v;
    }
    __syncthreads();

    const int wave = tid >> 5;
    const int lane = tid & 31;
    const int hi   = lane >> 4;   // half-wave select
    const int ln   = lane & 15;
    const int c0   = wave << 4;   // column block base

    v8f acc0 = {};
    v8f acc1 = {};

    // A 16x4 fragment layout: V0 = K0 (lanes 0-15) / K2 (lanes 16-31), V1 = K1/K3.
    // B 4x16 fragment layout: Vj lanes 0-15 = row k0+j, lanes 16-31 = row k0+2+j.
    #pragma unroll 4
    for (int k0 = 0; k0 < kDim; k0 += 4) {
        const int kk = k0 + 2 * hi;
        v2f a0, a1, b;
        a0[0] = sA[ln * kLda + kk];
        a0[1] = sA[ln * kLda + kk + 1];
        a1[0] = sA[(16 + ln) * kLda + kk];
        a1[1] = sA[(16 + ln) * kLda + kk + 1];
        b[0]  = W[(size_t)kk * kDim + c0 + ln];
        b[1]  = W[(size_t)(kk + 1) * kDim + c0 + ln];
        acc0 = __builtin_amdgcn_wmma_f32_16x16x4_f32(false, a0, false, b,
                                                     (short)0, acc0, false, false);
        acc1 = __builtin_amdgcn_wmma_f32_16x16x4_f32(false, a1, false, b,
                                                     (short)0, acc1, false, false);
    }

    // C/D layout: VGPR v, lanes 0-15 -> row v, lanes 16-31 -> row v+8; col = ln.
    const int colb = c0 + ln;
    const float bs = bias[colb];
    float* orow0 = Out + (size_t)(r0 + hi * 8) * kDim + colb;        // rows tile 0
    float* orow1 = orow0 + (size_t)16 * kDim;                        // rows tile 1
    float s = 0.f, ss = 0.f;
    #pragma unroll
    for (int v = 0; v < 8; ++v) {
        float x0 = acc0[v] + bs;
        float x1 = acc1[v] + bs;
        orow0[(size_t)v * kDim] = x0;
        orow1[(size_t)v * kDim] = x1;
        s  += x0 + x1;
        ss += x0 * x0 + x1 * x1;
    }
    atomicAdd(&colsum[colb], s);
    atomicAdd(&colsumsq[colb], ss);
}

// ---------------------------------------------------------------------------
// Fold BN stats + affine params into per-column scale/shift.
// mean = sum/N ; var = sumsq/N - mean^2 (biased, matches jnp.var)
// ---------------------------------------------------------------------------
__global__ void gin_bn_finalize(const float* __restrict__ colsum,
                                const float* __restrict__ colsumsq,
                                const float* __restrict__ g,
                                const float* __restrict__ b,
                                float* __restrict__ scale,
                                float* __restrict__ shift, float inv_n) {
    int c = threadIdx.x;
    if (c < kDim) {
        float m   = colsum[c] * inv_n;
        float var = colsumsq[c] * inv_n - m * m;
        float sc  = g[c] * rsqrtf(var + kEps);
        scale[c]  = sc;
        shift[c]  = b[c] - m * sc;
    }
}

// ---------------------------------------------------------------------------
// Fused BN-apply + ReLU, in place, float4 vectorized.
// ---------------------------------------------------------------------------
__global__ __launch_bounds__(256)
void gin_bn_relu(float* __restrict__ h, const float* __restrict__ scale,
                 const float* __restrict__ shift, int n4) {
    int i = blockIdx.x * 256 + threadIdx.x;
    if (i >= n4) return;
    int c4 = (i & 31) << 2;   // 128 cols = 32 float4 per row
    float4 v = ((float4*)h)[i];
    v.x = fmaxf(v.x * scale[c4 + 0] + shift[c4 + 0], 0.f);
    v.y = fmaxf(v.y * scale[c4 + 1] + shift[c4 + 1], 0.f);
    v.z = fmaxf(v.z * scale[c4 + 2] + shift[c4 + 2], 0.f);
    v.w = fmaxf(v.w * scale[c4 + 3] + shift[c4 + 3], 0.f);
    ((float4*)h)[i] = v;
}

// ---------------------------------------------------------------------------
// Tiny FC head: out[g, c] += pooled[g,:] @ w[:,c] + b[c]   (512 x 10)
// ---------------------------------------------------------------------------
__global__ __launch_bounds__(256)
void gin_fc(const float* __restrict__ pooled, const float* __restrict__ w,
            const float* __restrict__ b, float* __restrict__ out) {
    int t = blockIdx.x * 256 + threadIdx.x;
    if (t >= kGraphs * kClasses) return;
    int g = t / kClasses;
    int c = t % kClasses;
    float acc = b[c];
    #pragma unroll 8
    for (int k = 0; k < kDim; ++k)
        acc += pooled[(size_t)g * kDim + k] * w[(size_t)k * kClasses + c];
    out[t] += acc;
}

// ---------------------------------------------------------------------------
// In-place log_softmax over 10 classes per graph.
// ---------------------------------------------------------------------------
__global__ __launch_bounds__(256)
void gin_log_softmax(float* __restrict__ out) {
    int g = blockIdx.x * 256 + threadIdx.x;
    if (g >= kGraphs) return;
    float r[kClasses];
    float m = -3.402823466e+38f;
    #pragma unroll
    for (int c = 0; c < kClasses; ++c) {
        r[c] = out[g * kClasses + c];
        m = fmaxf(m, r[c]);
    }
    float sum = 0.f;
    #pragma unroll
    for (int c = 0; c < kClasses; ++c) sum += expf(r[c] - m);
    float lse = logf(sum) + m;
    #pragma unroll
    for (int c = 0; c < kClasses; ++c) out[g * kClasses + c] = r[c] - lse;
}

// ---------------------------------------------------------------------------
// Host-side orchestration
// ---------------------------------------------------------------------------
extern "C" void kernel_launch(void* const* d_in, const int* in_sizes, int n_in,
                              void* d_out, int out_size, void* d_ws, size_t ws_size,
                              hipStream_t stream) {
    (void)in_sizes; (void)n_in; (void)out_size; (void)ws_size;

    const float* x    = (const float*)d_in[0];
    const int*   eidx = (const int*)d_in[1];
    const int*   batch= (const int*)d_in[2];
    const float* w1   = (const float*)d_in[3];
    const float* b1   = (const float*)d_in[4];
    const float* bng1 = (const float*)d_in[5];
    const float* bnb1 = (const float*)d_in[6];
    const float* w2   = (const float*)d_in[7];
    const float* b2   = (const float*)d_in[8];
    const float* bng2 = (const float*)d_in[9];
    const float* bnb2 = (const float*)d_in[10];
    const float* fcw  = (const float*)d_in[11];
    const float* fcb  = (const float*)d_in[12];
    float* out = (float*)d_out;

    // Workspace layout (~103 MB): two ping-pong feature buffers + small scratch.
    float* ws       = (float*)d_ws;
    float* bufA     = ws;
    float* bufB     = bufA + (size_t)kNodes * kDim;
    float* pooled   = bufB + (size_t)kNodes * kDim;
    float* colsum   = pooled + (size_t)kGraphs * kDim;
    float* colsumsq = colsum + kDim;
    float* scale    = colsumsq + kDim;
    float* shift    = scale + kDim;

    const int* src = eidx;            // edge_index[0]
    const int* dst = eidx + kEdges;   // edge_index[1]

    const int n4         = kNodes * kDim / 4;            // 3.2M float4
    const int featBlocks = (n4 + 255) / 256;             // 12500
    const int scatBlocks = (kEdges * 32 + 255) / 256;    // 75000
    const int poolBlocks = (kNodes * 32 + 255) / 256;    // 12500
    const float inv_n    = 1.0f / (float)kNodes;

    // out accumulator = 0
    gin_fill<<<(kGraphs * kClasses + 255) / 256, 256, 0, stream>>>(out, 0.f, kGraphs * kClasses);

    // Layer-0 readout: pool(x) -> fc[0]
    gin_fill<<<(kGraphs * kDim + 255) / 256, 256, 0, stream>>>(pooled, 0.f, kGraphs * kDim);
    gin_pool<<<poolBlocks, 256, 0, stream>>>(x, batch, pooled);
    gin_fc<<<(kGraphs * kClasses + 255) / 256, 256, 0, stream>>>(pooled, fcw, fcb, out);

    const float* h = x;
    for (int i = 0; i < kLayers; ++i) {
        float* z = (i & 1) ? bufB : bufA;

        // z = h + scatter_add(h[src] -> dst)   (GIN aggregation, eps=0)
        gin_copy4<<<featBlocks, 256, 0, stream>>>((float4*)z, (const float4*)h, n4);
        gin_scatter_add<<<scatBlocks, 256, 0, stream>>>(h, src, dst, z);

        // z = ReLU(BN(z @ W1 + b1))   (in-place GEMM + fused stats)
        gin_fill<<<1, 256, 0, stream>>>(colsum, 0.f, 2 * kDim);
        gin_gemm_stats<<<kNodes / kBR, 256, 0, stream>>>(
            z, w1 + (size_t)i * kDim * kDim, b1 + i * kDim, z, colsum, colsumsq);
        gin_bn_finalize<<<1, kDim, 0, stream>>>(colsum, colsumsq,
            bng1 + i * kDim, bnb1 + i * kDim, scale, shift, inv_n);
        gin_bn_relu<<<featBlocks, 256, 0, stream>>>(z, scale, shift, n4);

        // z = ReLU(BN(z @ W2 + b2))
        gin_fill<<<1, 256, 0, stream>>>(colsum, 0.f, 2 * kDim);
        gin_gemm_stats<<<kNodes / kBR, 256, 0, stream>>>(
            z, w2 + (size_t)i * kDim * kDim, b2 + i * kDim, z, colsum, colsumsq);
        gin_bn_finalize<<<1, kDim, 0, stream>>>(colsum, colsumsq,
            bng2 + i * kDim, bnb2 + i * kDim, scale, shift, inv_n);
        gin_bn_relu<<<featBlocks, 256, 0, stream>>>(z, scale, shift, n4);

        // Readout for this layer: pool(z) -> fc[i+1]
        gin_fill<<<(kGraphs * kDim + 255) / 256, 256, 0, stream>>>(pooled, 0.f, kGraphs * kDim);
        gin_pool<<<poolBlocks, 256, 0, stream>>>(z, batch, pooled);
        gin_fc<<<(kGraphs * kClasses + 255) / 256, 256, 0, stream>>>(
            pooled, fcw + (size_t)(i + 1) * kDim * kClasses, fcb + (i + 1) * kClasses, out);

        h = z;
    }

    gin_log_softmax<<<(kGraphs + 255) / 256, 256, 0, stream>>>(out);
}